// DeformConv_89472758710284
// MI455X (gfx1250) — compile-verified
//
#include <hip/hip_runtime.h>

typedef __attribute__((ext_vector_type(2))) float v2f;
typedef __attribute__((ext_vector_type(4))) float v4f;
typedef __attribute__((ext_vector_type(8))) float v8f;

#define OHW   3136        // 56*56
#define HH    56
#define WW    56
#define OW    56
#define CIN   64
#define OCN   64
#define NPIX  64          // output pixels per block
#define CST   68          // c-dim stride for wl/vs: lanes 0-15 banks {4l,4l+1}, 16-31 {4l+2,4l+3}

// ---- pass 1: NCHW -> NHWC transpose of x into workspace
__global__ __launch_bounds__(256) void nchw_to_nhwc(
    const float* __restrict__ x, float* __restrict__ xT)
{
    __shared__ float t[64 * 65];
    const int tid = threadIdx.x;
    const int b   = blockIdx.x / 49;
    const int p0  = (blockIdx.x % 49) * 64;

    const float* xb = x + (long)b * CIN * OHW;
    {
        int pixl = tid & 63;                 // coalesced along pixels
        int r    = tid >> 6;
        for (int j = 0; j < 16; ++j) {
            int c = r + j * 4;
            t[c * 65 + pixl] = xb[(long)c * OHW + p0 + pixl];
        }
    }
    __syncthreads();
    {
        int c = tid & 63;                    // coalesced along channels
        int r = tid >> 6;
        float* o = xT + ((long)b * OHW + p0) * 64;
        for (int j = 0; j < 16; ++j) {
            int pixl = r + j * 4;
            o[(long)pixl * 64 + c] = t[c * 65 + pixl];
        }
    }
}

// ---- pass 2: fused deformable sampling + implicit GEMM via f32 WMMA
__global__ __launch_bounds__(256) void deform_conv_wmma_f32(
    const float* __restrict__ xT,
    const float* __restrict__ off,
    const float* __restrict__ wgt,
    float* __restrict__ out)
{
    __shared__ float wl[64 * CST];   // wl[oc][c]  weights for current tap (17408 B)
    __shared__ float vs[64 * CST];   // vs[p][c]   bilinear samples        (17408 B)

    const int tid  = threadIdx.x;
    const int lane = tid & 31;
    const int wave = tid >> 5;
    const int half = lane >> 4;
    const int l15  = lane & 15;

    const int b  = blockIdx.x / 49;
    const int p0 = (blockIdx.x % 49) * 64;

    const int oc_base = (wave & 3) * 16;     // M tile
    const int n0_base = (wave >> 2) * 32;    // two N tiles per wave
    const int n1_base = n0_base + 16;

    // sampling: one pixel per 16-lane group, 4 channels per lane, 4 passes
    const int sp = tid >> 4;                 // pixel-in-pass 0..15
    const int c4 = (tid & 15) * 4;           // channel quad

    v8f acc0 = {0.f, 0.f, 0.f, 0.f, 0.f, 0.f, 0.f, 0.f};
    v8f acc1 = {0.f, 0.f, 0.f, 0.f, 0.f, 0.f, 0.f, 0.f};

    const float* xTb  = xT  + (long)b * OHW * 64;
    const float* offb = off + (long)b * 18 * OHW;

    for (int k = 0; k < 9; ++k) {
        __syncthreads();  // protect LDS against previous tap's WMMA reads

        // ---- stage tap-k weights: wl[oc][c] = wgt[oc][c*9 + k]  (lane-contiguous in c)
        for (int j = 0; j < 16; ++j) {
            int e  = tid + j * 256;          // 0..4095
            int c  = e & 63;
            int oc = e >> 6;
            wl[oc * CST + c] = wgt[(long)oc * (CIN * 9) + c * 9 + k];
        }

        const int kdy = k / 3 - 1;
        const int kdx = k % 3 - 1;

        // ---- bilinear sampling: coalesced b128 channel-runs from NHWC
        for (int pass = 0; pass < 4; ++pass) {
            int p   = pass * 16 + sp;
            int pix = p0 + p;
            int yy  = pix / OW;
            int xx  = pix - yy * OW;
            float dy = offb[(long)(2 * k) * OHW + pix];
            float dx = offb[(long)(2 * k + 1) * OHW + pix];
            float py = (float)(yy + kdy) + dy;
            float px = (float)(xx + kdx) + dx;
            float fy0 = floorf(py), fx0 = floorf(px);
            int   y0 = (int)fy0,   x0 = (int)fx0;
            float ty = py - fy0,   tx = px - fx0;
            float w00 = (1.f - ty) * (1.f - tx);
            float w01 = (1.f - ty) * tx;
            float w10 = ty * (1.f - tx);
            float w11 = ty * tx;
            bool vy0 = (unsigned)y0       < (unsigned)HH;
            bool vy1 = (unsigned)(y0 + 1) < (unsigned)HH;
            bool vx0 = (unsigned)x0       < (unsigned)WW;
            bool vx1 = (unsigned)(x0 + 1) < (unsigned)WW;
            const float* base = xTb + ((long)y0 * WW + x0) * 64 + c4;
            v4f z = {0.f, 0.f, 0.f, 0.f};
            v4f s00 = (vy0 & vx0) ? *(const v4f*)(base)                : z;
            v4f s01 = (vy0 & vx1) ? *(const v4f*)(base + 64)           : z;
            v4f s10 = (vy1 & vx0) ? *(const v4f*)(base + WW * 64)      : z;
            v4f s11 = (vy1 & vx1) ? *(const v4f*)(base + WW * 64 + 64) : z;
            v4f v = s00 * w00 + s01 * w01 + s10 * w10 + s11 * w11;
            *(v4f*)&vs[p * CST + c4] = v;    // all 64 banks hit exactly once
        }

        __syncthreads();

        // ---- preload this tap's 16 A-fragments into registers
        v2f areg[16];
        #pragma unroll
        for (int kk = 0; kk < 16; ++kk) {
            int r = kk * 4 + 2 * half;       // even c index
            areg[kk] = *(const v2f*)&wl[(oc_base + l15) * CST + r];
        }

        // ---- 32 WMMA steps over the 64-channel K slice (2 N tiles share A)
        #pragma unroll
        for (int kk = 0; kk < 16; ++kk) {
            int r = kk * 4 + 2 * half;
            v2f b0 = *(const v2f*)&vs[(n0_base + l15) * CST + r];
            v2f b1 = *(const v2f*)&vs[(n1_base + l15) * CST + r];
            acc0 = __builtin_amdgcn_wmma_f32_16x16x4_f32(
                false, areg[kk], false, b0, (short)0, acc0, false, false);
            acc1 = __builtin_amdgcn_wmma_f32_16x16x4_f32(
                false, areg[kk], false, b1, (short)0, acc1, false, false);
        }
    }

    // ---- store D: VGPR r -> M = oc_base + r + 8*half, N = n_base + l15
    {
        long outb = (long)(b * OCN) * OHW + p0;
        #pragma unroll
        for (int r = 0; r < 8; ++r) {
            int m = oc_base + r + half * 8;
            out[outb + (long)m * OHW + (n0_base + l15)] = acc0[r];
            out[outb + (long)m * OHW + (n1_base + l15)] = acc1[r];
        }
    }
}

extern "C" void kernel_launch(void* const* d_in, const int* in_sizes, int n_in,
                              void* d_out, int out_size, void* d_ws, size_t ws_size,
                              hipStream_t stream) {
    (void)in_sizes; (void)n_in; (void)ws_size; (void)out_size;
    const float* x   = (const float*)d_in[0];
    const float* off = (const float*)d_in[1];
    const float* wgt = (const float*)d_in[2];
    float* out = (float*)d_out;
    float* xT  = (float*)d_ws;               // needs 8*3136*64*4 = 6.42 MB

    dim3 grid(8 * 49), block(256);
    nchw_to_nhwc<<<grid, block, 0, stream>>>(x, xT);
    deform_conv_wmma_f32<<<grid, block, 0, stream>>>(xT, off, wgt, out);
}